// GridGCN_42056319762623
// MI455X (gfx1250) — compile-verified
//
#include <hip/hip_runtime.h>

#define N_NODES 100000   // multiple of 16 -> no partial WMMA row tiles
#define N_EDGES 3200000
#define F_IN 128
#define H1 64
#define H2 32
#define G_GR 32

typedef __attribute__((ext_vector_type(2))) float v2f;
typedef __attribute__((ext_vector_type(8))) float v8f;

// ---------------- fill ----------------
__global__ void k_fill(float* __restrict__ p, float v, long long n) {
  long long i = (long long)blockIdx.x * blockDim.x + threadIdx.x;
  if (i < n) p[i] = v;
}

// ---------------- degree / dinv ----------------
__global__ void k_deg(const int* __restrict__ col, const float* __restrict__ ew,
                      float* __restrict__ deg, int ne) {
  int e = blockIdx.x * blockDim.x + threadIdx.x;
  if (e < ne) unsafeAtomicAdd(&deg[col[e]], ew[e]);
}

__global__ void k_dinv(const float* __restrict__ deg, float* __restrict__ dinv, int n) {
  int i = blockIdx.x * blockDim.x + threadIdx.x;
  if (i < n) dinv[i] = rsqrtf(deg[i]);
}

// ---------------- WMMA f32 GEMM: C[nrows x H] = A'[nrows x K] * B[K x H] ----------------
// One 16x16 C tile per wave32. B staged in LDS. Optional per-column input
// normalization folded into a single FMA per element:
//   a' = a * (1/std) + (-mean/std)
// Requires nrows % 16 == 0 (guaranteed by launcher: N_NODES = 100000 = 6250*16).
template <int K, int H, bool NORM>
__global__ void k_wmma_gemm(const float* __restrict__ A, const float* __restrict__ B,
                            const float* __restrict__ mean, const float* __restrict__ stdv,
                            float* __restrict__ C, int nrows) {
  __shared__ float Bs[K * H];
  __shared__ float Sc[NORM ? K : 1];  // 1/std
  __shared__ float Of[NORM ? K : 1];  // -mean/std

  for (int i = threadIdx.x; i < K * H; i += blockDim.x) Bs[i] = B[i];
  if (NORM) {
    for (int i = threadIdx.x; i < K; i += blockDim.x) {
      float rs = 1.0f / stdv[i];   // K divides per block, amortized
      Sc[i] = rs;
      Of[i] = -mean[i] * rs;
    }
  }
  __syncthreads();

  const int lane = threadIdx.x & 31;
  const int wave = threadIdx.x >> 5;
  const int colTiles = H / 16;
  const int rowTiles = nrows >> 4;

  int tile = blockIdx.x * (blockDim.x >> 5) + wave;
  int rowTile = tile / colTiles;
  int colTile = tile - rowTile * colTiles;
  if (rowTile >= rowTiles) return;  // wave-uniform exit; EXEC all-1 at every WMMA

  // ISA lane layout for 16x16x4 f32:
  //   A: lane holds rows m = lane&15, K-pair koff = 2*(lane>>4)
  //   B: lane holds col n = lane&15, same K-pair striping
  const int m    = rowTile * 16 + (lane & 15);
  const int koff = (lane >> 4) * 2;
  const int n    = colTile * 16 + (lane & 15);
  const float* __restrict__ arow = A + (size_t)m * K + koff;  // 8B-aligned

  v8f acc = {};
#pragma unroll
  for (int k = 0; k < K; k += 4) {
    v2f a = *(const v2f*)(arow + k);  // global_load_b64
    if (NORM) {
      a.x = fmaf(a.x, Sc[k + koff], Of[k + koff]);
      a.y = fmaf(a.y, Sc[k + koff + 1], Of[k + koff + 1]);
    }
    v2f b;
    b.x = Bs[(k + koff) * H + n];
    b.y = Bs[(k + koff + 1) * H + n];
    acc = __builtin_amdgcn_wmma_f32_16x16x4_f32(
        /*neg_a=*/false, a, /*neg_b=*/false, b,
        /*c_mod=*/(short)0, acc, /*reuse_a=*/false, /*reuse_b=*/false);
  }

  // C/D layout: VGPR r -> row r (lanes 0-15) or r+8 (lanes 16-31), col = lane&15
  const int crowBase = rowTile * 16 + (lane >> 4) * 8;
#pragma unroll
  for (int r = 0; r < 8; ++r) {
    C[(size_t)(crowBase + r) * H + n] = acc[r];  // no guard: nrows % 16 == 0
  }
}

// ---------------- edge aggregation: agg[col] += h[row] * (dinv[row]*dinv[col]*ew) ----------
template <int F>
__global__ void k_edge_agg(const int* __restrict__ row, const int* __restrict__ col,
                           const float* __restrict__ ew, const float* __restrict__ dinv,
                           const float* __restrict__ h, float* __restrict__ agg, int ne) {
  long long tid = (long long)blockIdx.x * blockDim.x + threadIdx.x;
  long long e = tid >> (F == 64 ? 6 : 5);
  int f = (int)(tid & (F - 1));
  if (e >= ne) return;
  int r = row[e], c = col[e];
  float coeff = dinv[r] * dinv[c] * ew[e];
  unsafeAtomicAdd(&agg[(size_t)c * F + f], h[(size_t)r * F + f] * coeff);
}

// ---------------- combine: out = relu(agg + dinv^2 * hlin + b), in place over agg ---------
template <int F>
__global__ void k_combine(float* __restrict__ agg, const float* __restrict__ hlin,
                          const float* __restrict__ dinv, const float* __restrict__ bias,
                          int n) {
  long long tid = (long long)blockIdx.x * blockDim.x + threadIdx.x;
  if (tid >= (long long)n * F) return;
  int i = (int)(tid >> (F == 64 ? 6 : 5));
  int f = (int)(tid & (F - 1));
  float d = dinv[i];
  float v = agg[tid] + d * d * hlin[tid] + bias[f];
  agg[tid] = v > 0.0f ? v : 0.0f;
}

// ---------------- pooling ----------------
__global__ void k_count(const int* __restrict__ batch, float* __restrict__ cnt, int n) {
  int i = blockIdx.x * blockDim.x + threadIdx.x;
  if (i < n) unsafeAtomicAdd(&cnt[batch[i]], 1.0f);
}

__global__ void k_pool(const float* __restrict__ h2, const int* __restrict__ batch,
                       float* __restrict__ s, int n) {
  long long tid = (long long)blockIdx.x * blockDim.x + threadIdx.x;
  if (tid >= (long long)n * H2) return;
  int i = (int)(tid >> 5);
  int f = (int)(tid & 31);
  unsafeAtomicAdd(&s[(size_t)batch[i] * H2 + f], h2[tid]);
}

// ---------------- readout MLP: g=[mean, sum] -> relu(g@Wr1+br1)@Wr2+br2 ----------------
__global__ void k_readout(const float* __restrict__ s, const float* __restrict__ cnt,
                          const float* __restrict__ Wr1, const float* __restrict__ br1,
                          const float* __restrict__ Wr2, const float* __restrict__ br2,
                          float* __restrict__ out) {
  int g = threadIdx.x;
  if (g >= G_GR) return;
  float feat[2 * H2];
  float c = cnt[g];
  c = c > 1.0f ? c : 1.0f;
  float rc = 1.0f / c;
#pragma unroll
  for (int j = 0; j < H2; ++j) {
    float sv = s[g * H2 + j];
    feat[j]      = sv * rc;  // mean pool
    feat[H2 + j] = sv;       // add pool
  }
  float o = br2[0];
  for (int j = 0; j < H2; ++j) {
    float acc = br1[j];
#pragma unroll
    for (int k = 0; k < 2 * H2; ++k) acc += feat[k] * Wr1[k * H2 + j];
    acc = acc > 0.0f ? acc : 0.0f;
    o += acc * Wr2[j];
  }
  out[g] = o;
}

// ---------------- host launcher ----------------
extern "C" void kernel_launch(void* const* d_in, const int* in_sizes, int n_in,
                              void* d_out, int out_size, void* d_ws, size_t ws_size,
                              hipStream_t stream) {
  (void)in_sizes; (void)n_in; (void)out_size; (void)ws_size;
  const float* x    = (const float*)d_in[0];
  const int*   eidx = (const int*)d_in[1];
  const float* ew   = (const float*)d_in[2];
  const int*   bat  = (const int*)d_in[3];
  const float* fm   = (const float*)d_in[4];
  const float* fs   = (const float*)d_in[5];
  const float* W1   = (const float*)d_in[6];
  const float* b1   = (const float*)d_in[7];
  const float* W2   = (const float*)d_in[8];
  const float* b2   = (const float*)d_in[9];
  const float* Wr1  = (const float*)d_in[10];
  const float* br1  = (const float*)d_in[11];
  const float* Wr2  = (const float*)d_in[12];
  const float* br2  = (const float*)d_in[13];
  float* out = (float*)d_out;

  const int* erow = eidx;            // edge_index[0]
  const int* ecol = eidx + N_EDGES;  // edge_index[1]

  float* ws   = (float*)d_ws;
  float* hlin = ws;                          // N*64 (layer1) / N*32 (layer2 reuse)
  float* h1   = hlin + (size_t)N_NODES * H1; // agg1, then relu output in-place
  float* h2   = h1 + (size_t)N_NODES * H1;   // agg2, then relu output in-place
  float* deg  = h2 + (size_t)N_NODES * H2;   // N
  float* dinv = deg + N_NODES;               // N
  float* spool = dinv + N_NODES;             // G*32 sums
  float* cnt   = spool + G_GR * H2;          // G

  const int TB = 256;
  auto blk = [](long long n, int tb) { return (int)((n + tb - 1) / tb); };

  // init scratch (ws is poisoned; must self-initialize every call)
  k_fill<<<blk(N_NODES, TB), TB, 0, stream>>>(deg, 1.0f, N_NODES);               // self-loop
  k_fill<<<blk((long long)N_NODES * H1, TB), TB, 0, stream>>>(h1, 0.0f, (long long)N_NODES * H1);
  k_fill<<<blk((long long)N_NODES * H2, TB), TB, 0, stream>>>(h2, 0.0f, (long long)N_NODES * H2);
  k_fill<<<1, TB, 0, stream>>>(spool, 0.0f, G_GR * H2 + G_GR);                   // sums + cnt

  // degrees -> dinv
  k_deg<<<blk(N_EDGES, TB), TB, 0, stream>>>(ecol, ew, deg, N_EDGES);
  k_dinv<<<blk(N_NODES, TB), TB, 0, stream>>>(deg, dinv, N_NODES);

  // ---- layer 1: hlin = ((x-mean)/std) @ W1  [N,128]x[128,64] via WMMA f32 ----
  {
    int tiles = (N_NODES / 16) * (H1 / 16);
    k_wmma_gemm<F_IN, H1, true><<<blk(tiles, 8), TB, 0, stream>>>(x, W1, fm, fs, hlin, N_NODES);
  }
  k_edge_agg<H1><<<blk((long long)N_EDGES * H1, TB), TB, 0, stream>>>(erow, ecol, ew, dinv, hlin, h1, N_EDGES);
  k_combine<H1><<<blk((long long)N_NODES * H1, TB), TB, 0, stream>>>(h1, hlin, dinv, b1, N_NODES);

  // ---- layer 2: hlin2 = h1 @ W2  [N,64]x[64,32] via WMMA f32 (reuse hlin buf) ----
  {
    int tiles = (N_NODES / 16) * (H2 / 16);
    k_wmma_gemm<H1, H2, false><<<blk(tiles, 8), TB, 0, stream>>>(h1, W2, nullptr, nullptr, hlin, N_NODES);
  }
  k_edge_agg<H2><<<blk((long long)N_EDGES * H2, TB), TB, 0, stream>>>(erow, ecol, ew, dinv, hlin, h2, N_EDGES);
  k_combine<H2><<<blk((long long)N_NODES * H2, TB), TB, 0, stream>>>(h2, hlin, dinv, b2, N_NODES);

  // ---- pooling + readout ----
  k_count<<<blk(N_NODES, TB), TB, 0, stream>>>(bat, cnt, N_NODES);
  k_pool<<<blk((long long)N_NODES * H2, TB), TB, 0, stream>>>(h2, bat, spool, N_NODES);
  k_readout<<<1, 32, 0, stream>>>(spool, cnt, Wr1, br1, Wr2, br2, out);
}